// AttentionBlock_54400055771902
// MI455X (gfx1250) — compile-verified
//
#include <hip/hip_runtime.h>
#include <hip/hip_bf16.h>
#include <stdint.h>

// ---------------------------------------------------------------------------
// GAT edge-attention layer for MI455X (gfx1250, wave32, WMMA).
//
// Roofline: all GEMMs total ~6.2 GFLOP (negligible vs fp32 WMMA rate); the
// kernel is bound by edge-stream traffic (~0.5-1 GB @ 23.3 TB/s). Node arrays
// (~15 MB) stay resident in the 192 MB L2, so Q/K/V gathers are L2 hits.
// The 640000x32 @ 32x128 edge GEMM is fused into the logit pass so the
// 327 MB edge projection never round-trips HBM.
//
// New this round: weight matrices (W: 64KB, W_E: 16KB) are staged into LDS
// once per workgroup via GLOBAL_LOAD_ASYNC_TO_LDS_B128 (ASYNCcnt-tracked
// async copy, CDNA5 ISA ch.10.7/15.18) + s_wait_asynccnt, so WMMA B-operands
// come from conflict-free ds_load instead of per-wave strided global loads.
// ---------------------------------------------------------------------------

typedef float v2f __attribute__((ext_vector_type(2)));
typedef float v8f __attribute__((ext_vector_type(8)));

#define N_NODES   10000
#define N_EDGES   640000
#define CHANNELS  128
#define N_HEADS   8
#define HEAD_DIM  16
#define BASIS_DIM 32
#define NEG_SLOPE 0.01f

// ---- CDNA5 async global->LDS copy helpers (inline asm; ASYNCcnt-tracked) ---
// LDS address is taken from the VDST VGPR: the low 32 bits of a flat LDS pointer are
// the LDS byte offset (ISA §10.2 aperture rules), so truncation is correct.
__device__ __forceinline__ void async_copy_b128(void* lds_dst, const void* gsrc) {
    uint32_t ldsoff = (uint32_t)(uintptr_t)lds_dst;
    uint64_t gaddr  = (uint64_t)(uintptr_t)gsrc;
    asm volatile("global_load_async_to_lds_b128 %0, %1, off"
                 :: "v"(ldsoff), "v"(gaddr) : "memory");
}
__device__ __forceinline__ void async_wait0() {
    asm volatile("s_wait_asynccnt 0" ::: "memory");
}

// CAS-based float atomic max (portable; values are finite logits, init -1e30)
__device__ __forceinline__ void atomic_max_f32(float* addr, float val) {
    unsigned int* ua = (unsigned int*)addr;
    unsigned int old = *ua;
    while (true) {
        float cur = __uint_as_float(old);
        if (cur >= val) break;
        unsigned int assumed = old;
        old = atomicCAS(ua, assumed, __float_as_uint(val));
        if (old == assumed) break;
    }
}

// ---------------------------------------------------------------------------
// 0) Init: harness poisons d_out/d_ws; zero/neg-inf everything we accumulate.
// ---------------------------------------------------------------------------
__global__ void init_kernel(float* __restrict__ out,
                            float* __restrict__ seg_max,
                            float* __restrict__ seg_sum) {
    const int i = blockIdx.x * blockDim.x + threadIdx.x;
    if (i < N_NODES * CHANNELS) out[i] = 0.0f;
    if (i < N_NODES * N_HEADS) {
        seg_max[i] = -1e30f;
        seg_sum[i] = 0.0f;
    }
}

// ---------------------------------------------------------------------------
// 1) Node projection: Out[10000x128] = A[10000x128] @ W[128x128] via
//    V_WMMA_F32_16X16X4_F32. One wave per 16x16 tile, 32 WMMA ops (K=128).
//    W staged into 64KB LDS per block by async copies; B reads are ds_load.
//    A 16x4 layout: lanes 0-15 hold M=lane,(K=k0,k0+1); lanes 16-31 (k0+2,k0+3).
//    B 4x16 layout: VGPR j, lanes 0-15 row k0+2j, lanes 16-31 row k0+2j+1.
// ---------------------------------------------------------------------------
__global__ void node_proj_wmma(const float* __restrict__ A,
                               const float* __restrict__ W,
                               float* __restrict__ Out) {
    __shared__ float sW[CHANNELS * CHANNELS];   // 64 KB of the 320 KB WGP LDS

    // Async-stage W: 16384 dwords = 4096 b128 copies, 32 per thread (128 thr).
    for (int i = threadIdx.x; i < (CHANNELS * CHANNELS) / 4; i += blockDim.x)
        async_copy_b128(sW + i * 4, W + i * 4);
    async_wait0();          // this wave's async copies complete
    __syncthreads();        // all waves' copies visible

    const int wave = blockIdx.x * (blockDim.x >> 5) + (threadIdx.x >> 5);
    const int lane = threadIdx.x & 31;
    const int half = lane >> 4;   // 0: lanes 0-15, 1: lanes 16-31
    const int l16  = lane & 15;
    const int mt   = wave >> 3;   // 625 M tiles (10000 = 625*16, exact)
    const int nt   = wave & 7;    // 8 N tiles  (128 = 8*16)
    const int m    = mt * 16 + l16;
    const int n    = nt * 16 + l16;

    v8f acc = {};
    const float* arow = A + (size_t)m * CHANNELS;
#pragma unroll 8
    for (int k0 = 0; k0 < CHANNELS; k0 += 4) {
        v2f a = *(const v2f*)(arow + k0 + 2 * half);
        v2f b;
        b.x = sW[(k0 + half) * CHANNELS + n];        // ds_load, conflict-free
        b.y = sW[(k0 + 2 + half) * CHANNELS + n];
        acc = __builtin_amdgcn_wmma_f32_16x16x4_f32(
            /*neg_a=*/false, a, /*neg_b=*/false, b,
            /*c_mod=*/(short)0, acc, /*reuse_a=*/false, /*reuse_b=*/false);
    }
    // D layout: VGPR r -> row mt*16 + r + 8*half, col n
#pragma unroll
    for (int r = 0; r < 8; ++r)
        Out[(size_t)(mt * 16 + half * 8 + r) * CHANNELS + n] = acc[r];
}

// ---------------------------------------------------------------------------
// 2) Fused edge pass: one wave per 16 edges.
//    a) Eproj(16x128) = edge_attrs(16x32) @ W_E(32x128): 8 N-tiles x 8 K-steps
//       = 64 fp32 WMMA ops, result kept in registers (8 x v8f).
//       W_E staged into 16KB LDS per block via async copies.
//    b) val = Eproj + Q[dst] + K[src]; leaky-ReLU; dot with attention[head].
//       N-tile t == head t (HEAD_DIM == 16). 16-lane shfl_xor reduction.
//    c) store logits[E,8]; atomic-max into seg_max[dst,head].
// ---------------------------------------------------------------------------
__global__ void edge_logits_wmma(const float* __restrict__ edge_attrs,
                                 const int*   __restrict__ eidx,      // [2, E]
                                 const float* __restrict__ W_E,
                                 const float* __restrict__ att,       // [8,16]
                                 const float* __restrict__ Qn,
                                 const float* __restrict__ Kn,
                                 float* __restrict__ logits,          // [E,8]
                                 float* __restrict__ seg_max) {       // [N,8]
    __shared__ float sWE[BASIS_DIM * CHANNELS];  // 16 KB

    // Async-stage W_E: 4096 dwords = 1024 b128 copies, 4 per thread (256 thr).
    for (int i = threadIdx.x; i < (BASIS_DIM * CHANNELS) / 4; i += blockDim.x)
        async_copy_b128(sWE + i * 4, W_E + i * 4);
    async_wait0();
    __syncthreads();

    const int wave = blockIdx.x * (blockDim.x >> 5) + (threadIdx.x >> 5);
    const int lane = threadIdx.x & 31;
    const int half = lane >> 4;
    const int l16  = lane & 15;
    const int e0   = wave * 16;   // 640000 = 40000*16, exact

    // Prefetch the next wave's edge_attrs tile (speculative, OOB-safe).
    __builtin_prefetch(edge_attrs + (size_t)(e0 + 16 + l16) * BASIS_DIM, 0, 0);

    v8f acc[8] = {};
    const float* arow = edge_attrs + (size_t)(e0 + l16) * BASIS_DIM;
#pragma unroll
    for (int k0 = 0; k0 < BASIS_DIM; k0 += 4) {
        v2f a = *(const v2f*)(arow + k0 + 2 * half);
#pragma unroll
        for (int t = 0; t < 8; ++t) {
            const int n = t * 16 + l16;
            v2f b;
            b.x = sWE[(k0 + half) * CHANNELS + n];
            b.y = sWE[(k0 + 2 + half) * CHANNELS + n];
            acc[t] = __builtin_amdgcn_wmma_f32_16x16x4_f32(
                false, a, false, b, (short)0, acc[t], false, false);
        }
    }

    float att_v[8];
#pragma unroll
    for (int t = 0; t < 8; ++t) att_v[t] = att[t * 16 + l16];

#pragma unroll
    for (int r = 0; r < 8; ++r) {
        // This lane's C/D element in VGPR r belongs to edge e0 + r + 8*half.
        const int e    = e0 + half * 8 + r;
        const int srce = eidx[e];
        const int dste = eidx[N_EDGES + e];
        const float* qrow = Qn + (size_t)dste * CHANNELS;  // L2-resident gather
        const float* krow = Kn + (size_t)srce * CHANNELS;
#pragma unroll
        for (int t = 0; t < 8; ++t) {
            const int c = t * 16 + l16;           // head t, dim l16
            float v = acc[t][r] + qrow[c] + krow[c];
            v = v > 0.0f ? v : NEG_SLOPE * v;     // leaky ReLU
            float p = att_v[t] * v;
            // Sum over the 16 head dims: masks 1/2/4/8 stay within each half.
            p += __shfl_xor(p, 1);
            p += __shfl_xor(p, 2);
            p += __shfl_xor(p, 4);
            p += __shfl_xor(p, 8);
            if (l16 == 0) {                       // lanes 0 and 16 write
                logits[(size_t)e * N_HEADS + t] = p;
                atomic_max_f32(&seg_max[dste * N_HEADS + t], p);
            }
        }
    }
}

// ---------------------------------------------------------------------------
// 3) ex = exp(logit - seg_max[dst]); accumulate seg_sum[dst].
// ---------------------------------------------------------------------------
__global__ void edge_exp_kernel(const float* __restrict__ logits,
                                const int*   __restrict__ eidx,
                                const float* __restrict__ seg_max,
                                float* __restrict__ ex,
                                float* __restrict__ seg_sum) {
    const int i = blockIdx.x * blockDim.x + threadIdx.x;
    if (i >= N_EDGES * N_HEADS) return;
    const int e = i >> 3;
    const int h = i & 7;
    const int dste = eidx[N_EDGES + e];
    const float v = __expf(logits[i] - seg_max[dste * N_HEADS + h]);
    ex[i] = v;
    atomicAdd(&seg_sum[dste * N_HEADS + h], v);
}

// ---------------------------------------------------------------------------
// 4) Weighted scatter-add: one wave per edge; lane L -> channels 4L..4L+3
//    (head = L>>2). float4 V loads, global_atomic_add_f32 into d_out.
// ---------------------------------------------------------------------------
__global__ void scatter_out_kernel(const float* __restrict__ ex,
                                   const float* __restrict__ seg_sum,
                                   const int*   __restrict__ eidx,
                                   const float* __restrict__ Vn,
                                   float* __restrict__ out) {
    const int e    = blockIdx.x * (blockDim.x >> 5) + (threadIdx.x >> 5);
    const int lane = threadIdx.x & 31;
    if (e >= N_EDGES) return;
    const int srce = eidx[e];
    const int dste = eidx[N_EDGES + e];
    const int h    = lane >> 2;
    const int c0   = lane * 4;
    const float w  = ex[(size_t)e * N_HEADS + h] / seg_sum[dste * N_HEADS + h];
    const float4 v = *(const float4*)(Vn + (size_t)srce * CHANNELS + c0);
    float* o = out + (size_t)dste * CHANNELS + c0;
    atomicAdd(o + 0, w * v.x);
    atomicAdd(o + 1, w * v.y);
    atomicAdd(o + 2, w * v.z);
    atomicAdd(o + 3, w * v.w);
}

// ---------------------------------------------------------------------------
extern "C" void kernel_launch(void* const* d_in, const int* in_sizes, int n_in,
                              void* d_out, int out_size, void* d_ws, size_t ws_size,
                              hipStream_t stream) {
    const float* senders    = (const float*)d_in[0];
    const float* receivers  = (const float*)d_in[1];
    const int*   eidx       = (const int*)  d_in[2];
    const float* edge_attrs = (const float*)d_in[3];
    const float* W_Q        = (const float*)d_in[4];
    const float* W_K        = (const float*)d_in[5];
    const float* W_V        = (const float*)d_in[6];
    const float* W_E        = (const float*)d_in[7];
    const float* att        = (const float*)d_in[8];
    float* out = (float*)d_out;

    // Workspace layout (floats): ~57 MB total.
    float* ws      = (float*)d_ws;
    float* Qn      = ws;                                     // 10000*128
    float* Kn      = Qn + (size_t)N_NODES * CHANNELS;        // 10000*128
    float* Vn      = Kn + (size_t)N_NODES * CHANNELS;        // 10000*128
    float* logits  = Vn + (size_t)N_NODES * CHANNELS;        // 640000*8
    float* ex      = logits + (size_t)N_EDGES * N_HEADS;     // 640000*8
    float* seg_max = ex + (size_t)N_EDGES * N_HEADS;         // 10000*8
    float* seg_sum = seg_max + (size_t)N_NODES * N_HEADS;    // 10000*8

    // 0) init accumulators (covers max(N*C, N*H) elements)
    init_kernel<<<(N_NODES * CHANNELS + 255) / 256, 256, 0, stream>>>(out, seg_max, seg_sum);

    // 1) node projections: 625*8 = 5000 tiles, 4 waves/block -> 1250 blocks
    node_proj_wmma<<<1250, 128, 0, stream>>>(receivers, W_Q, Qn);
    node_proj_wmma<<<1250, 128, 0, stream>>>(senders,   W_K, Kn);
    node_proj_wmma<<<1250, 128, 0, stream>>>(senders,   W_V, Vn);

    // 2) fused edge GEMM + logits + seg_max: 40000 waves, 8 waves/block
    edge_logits_wmma<<<5000, 256, 0, stream>>>(edge_attrs, eidx, W_E, att,
                                               Qn, Kn, logits, seg_max);

    // 3) softmax numerator + denominator: 5.12M threads
    edge_exp_kernel<<<(N_EDGES * N_HEADS + 255) / 256, 256, 0, stream>>>(
        logits, eidx, seg_max, ex, seg_sum);

    // 4) weighted scatter-add: 640000 waves, 8 waves/block
    scatter_out_kernel<<<N_EDGES / 8, 256, 0, stream>>>(ex, seg_sum, eidx, Vn, out);
}